// RNNLayer_58299886076554
// MI455X (gfx1250) — compile-verified
//
#include <hip/hip_runtime.h>
#include <math.h>

// Problem sizes (fixed by the reference).
#define B_  32
#define L_  2048
#define IN_ 512
#define H_  512
#define C_  1000

// Persistent recurrence configuration.
#define NWG_          8      // workgroups in the persistent recurrence kernel
#define COLS_PER_WG_  64     // H_/NWG_
#define WAVES_PER_WG_ 4
#define COLS_PER_WAVE_ 16
#define LDS_PITCH_    516    // 512 + 4 pad floats -> conflict-free ds_load_b64

typedef float v2f __attribute__((ext_vector_type(2)));
typedef float v8f __attribute__((ext_vector_type(8)));

// CDNA5 fp32 tensor op: D(16x16,f32) = A(16x4,f32) x B(4x16,f32) + C
// A frag (2 VGPR): lane L holds A[M=L%16][K = 2*(L/16)+r], r=0,1 (contiguous in K)
// B frag (2 VGPR): lane L holds B[K = 2*(L/16)+r][N=L%16]   (contiguous in K for (N,K)-major W)
// C/D (8 VGPR):    reg v, lane L -> C[M = v + 8*(L/16)][N = L%16]
__device__ __forceinline__ v8f wmma4(v2f a, v2f b, v8f c) {
  return __builtin_amdgcn_wmma_f32_16x16x4_f32(
      /*neg_a=*/false, a, /*neg_b=*/false, b,
      /*c_mod=*/(short)0, c, /*reuse_a=*/false, /*reuse_b=*/false);
}

// ---------------------------------------------------------------------------
// Big parallel GEMMs: Y[m,n] = sum_k A[m*strideA + k] * W[n*K + k] + bias[n]
// Each wave computes a 32(M) x 64(N) strip: 2x4 tiles of 16x16.
// GUARD_N: clamp W-row pointer for n>=N (garbage accumulates into columns we
// never store) so EXEC stays all-ones around WMMA.
// ---------------------------------------------------------------------------
template <bool GUARD_N>
__global__ void gemm_nt_bias(const float* __restrict__ A, long strideA,
                             const float* __restrict__ W,      // (N,K) row-major
                             const float* __restrict__ bias,   // (N)
                             float* __restrict__ Y, int ldy,
                             int M, int N, int K, int nblocks) {
  const int wave = blockIdx.x * (blockDim.x >> 5) + (threadIdx.x >> 5);
  const int lane = threadIdx.x & 31;
  const int lo = lane & 15;
  const int hi = lane >> 4;
  const int nblk = wave % nblocks;
  const int mblk = wave / nblocks;
  const int row0 = mblk * 32;
  const int col0 = nblk * 64;
  if (row0 >= M) return;  // uniform per wave

  v8f acc[2][4];
#pragma unroll
  for (int i = 0; i < 2; ++i)
#pragma unroll
    for (int j = 0; j < 4; ++j) acc[i][j] = (v8f){0, 0, 0, 0, 0, 0, 0, 0};

  const float* a0 = A + (long)(row0 + lo) * strideA + 2 * hi;
  const float* a1 = A + (long)(row0 + 16 + lo) * strideA + 2 * hi;
  const float* bp[4];
#pragma unroll
  for (int j = 0; j < 4; ++j) {
    int n = col0 + j * 16 + lo;
    if (GUARD_N && n >= N) n = 0;  // clamp keeps loads in-bounds, EXEC uniform
    bp[j] = W + (long)n * K + 2 * hi;
  }

#pragma unroll 2
  for (int k = 0; k < K; k += 4) {
    const v2f af0 = *(const v2f*)(a0 + k);
    const v2f af1 = *(const v2f*)(a1 + k);
    v2f bf[4];
#pragma unroll
    for (int j = 0; j < 4; ++j) bf[j] = *(const v2f*)(bp[j] + k);
#pragma unroll
    for (int j = 0; j < 4; ++j) {
      acc[0][j] = wmma4(af0, bf[j], acc[0][j]);
      acc[1][j] = wmma4(af1, bf[j], acc[1][j]);
    }
  }

#pragma unroll
  for (int j = 0; j < 4; ++j) {
    const int n = col0 + j * 16 + lo;
    if (GUARD_N && n >= N) continue;  // after WMMA: divergence is fine now
    const float bn = bias[n];
#pragma unroll
    for (int v = 0; v < 8; ++v) {
      const int r = row0 + v + 8 * hi;
      Y[(long)r * ldy + n] = acc[0][j][v] + bn;
      Y[(long)(r + 16) * ldy + n] = acc[1][j][v] + bn;
    }
  }
}

// ---------------------------------------------------------------------------
// Persistent recurrence kernel.
// 8 workgroups x 4 waves. Each workgroup stages its 64x512 slice of Wh into
// LDS once (pitch 516 floats -> all 32 lanes of a ds_load_b64 hit distinct
// bank pairs). Each wave owns 16 output columns. Per timestep:
//   h_t[b,n] = tanh( xp_t[b,n] + sum_k h_{t-1}[b,k]*Wh[n,k] + bh[n] )
// xp_t+1 is prefetched into registers BEFORE the cross-workgroup barrier so
// the barrier spin hides its latency (xp depends only on t, not on peers).
// Coherence note: each step stores/loads *fresh* addresses (hn + t*H), so no
// WGP ever re-reads an address it cached before the producing step.
// ---------------------------------------------------------------------------
__global__ void rnn_persistent(const float* __restrict__ h0,   // hidden[:,0,:], batch stride L*H
                               const float* __restrict__ xp,   // (B*L, H) scratch
                               const float* __restrict__ Wh,   // (H,H) row-major
                               const float* __restrict__ bh,   // (H)
                               float* __restrict__ hn,         // (B,L,H) output region
                               unsigned* __restrict__ cnt) {   // barrier counter (zeroed)
  extern __shared__ float lWh[];  // COLS_PER_WG_ * LDS_PITCH_ floats = 132 KB

  const int tid = threadIdx.x;
  const int wave = tid >> 5;
  const int lane = tid & 31;
  const int lo = lane & 15;
  const int hi = lane >> 4;
  const int col_base = blockIdx.x * COLS_PER_WG_;

  // Stage Wh slice into LDS (row pitch LDS_PITCH_), float4 copies.
  for (int idx = tid; idx < COLS_PER_WG_ * (H_ / 4); idx += blockDim.x) {
    const int r = idx / (H_ / 4);
    const int c4 = idx % (H_ / 4);
    const float4 v = ((const float4*)(Wh + (long)(col_base + r) * H_))[c4];
    *(float4*)&lWh[r * LDS_PITCH_ + c4 * 4] = v;
  }
  __syncthreads();

  const int nloc = wave * COLS_PER_WAVE_;  // this wave's local column offset
  const int ncol = col_base + nloc + lo;   // this lane's output column
  const float bn = bh[ncol];
  const long RS = (long)L_ * H_;           // batch stride of (B,L,*) slices

  const float* lb = &lWh[(nloc + lo) * LDS_PITCH_ + 2 * hi];  // B frag base (LDS)

  // Software-pipelined xp values: xv[v] = xp_t[b = v + 8*hi, ncol],
  // xv[v+8] = xp_t[b = v + 8*hi + 16, ncol]. Preload for t=0.
  float xv[16];
#pragma unroll
  for (int v = 0; v < 8; ++v) {
    xv[v] = xp[(long)(v + 8 * hi) * RS + ncol];
    xv[v + 8] = xp[(long)(v + 8 * hi + 16) * RS + ncol];
  }

  const float* hprev = h0;
  for (int t = 0; t < L_; ++t) {
    v8f acc0 = (v8f){0, 0, 0, 0, 0, 0, 0, 0};
    v8f acc1 = (v8f){0, 0, 0, 0, 0, 0, 0, 0};
    const float* a0 = hprev + (long)lo * RS + 2 * hi;
    const float* a1 = hprev + (long)(lo + 16) * RS + 2 * hi;
#pragma unroll 4
    for (int k = 0; k < H_; k += 4) {
      const v2f af0 = *(const v2f*)(a0 + k);
      const v2f af1 = *(const v2f*)(a1 + k);
      const v2f bf = *(const v2f*)(lb + k);   // ds_load_b64, conflict-free
      acc0 = wmma4(af0, bf, acc0);
      acc1 = wmma4(af1, bf, acc1);
    }

    // Epilogue: consume register-resident xp values; store h_t.
    float* ho = hn + (long)t * H_;
#pragma unroll
    for (int v = 0; v < 8; ++v) {
      const int b0 = v + 8 * hi;   // batch rows 0..15
      const int b1 = b0 + 16;      // batch rows 16..31
      ho[(long)b0 * RS + ncol] = tanhf(acc0[v] + xv[v] + bn);
      ho[(long)b1 * RS + ncol] = tanhf(acc1[v] + xv[v + 8] + bn);
    }

    // Prefetch next step's xp slice; latency hidden behind the barrier.
    {
      const int tn = (t + 1 < L_) ? (t + 1) : t;  // clamp (last value unused)
      const float* xt = xp + (long)tn * H_;
#pragma unroll
      for (int v = 0; v < 8; ++v) {
        xv[v] = xt[(long)(v + 8 * hi) * RS + ncol];
        xv[v + 8] = xt[(long)(v + 8 * hi + 16) * RS + ncol];
      }
    }

    // Cross-workgroup barrier: all NWG_ workgroups must finish step t.
    __syncthreads();
    __threadfence();
    if (tid == 0) {
      __hip_atomic_fetch_add(cnt, 1u, __ATOMIC_RELEASE, __HIP_MEMORY_SCOPE_AGENT);
      const unsigned target = (unsigned)(NWG_ * (t + 1));
      while (__hip_atomic_load(cnt, __ATOMIC_ACQUIRE, __HIP_MEMORY_SCOPE_AGENT) < target)
        __builtin_amdgcn_s_sleep(1);
    }
    __syncthreads();

    hprev = hn + (long)t * H_;
  }
}

__global__ void zero_counter(unsigned* c) { *c = 0; }

extern "C" void kernel_launch(void* const* d_in, const int* in_sizes, int n_in,
                              void* d_out, int out_size, void* d_ws, size_t ws_size,
                              hipStream_t stream) {
  const float* inputs = (const float*)d_in[0];  // (B,L,IN)
  const float* hidden = (const float*)d_in[1];  // (B,L,H); we use [:,0,:]
  const float* Wi = (const float*)d_in[2];      // (H,IN)
  const float* bi = (const float*)d_in[3];      // (H)
  const float* Wh = (const float*)d_in[4];      // (H,H)
  const float* bh = (const float*)d_in[5];      // (H)
  const float* Wo = (const float*)d_in[6];      // (C,H)
  const float* bo = (const float*)d_in[7];      // (C)

  float* out = (float*)d_out;                   // (B,L,C)
  float* hn = out + (long)B_ * L_ * C_;         // (B,L,H)
  float* xp = (float*)d_ws;                     // (B*L, H) fp32 scratch: 128 MB
  unsigned* cnt = (unsigned*)(xp + (long)B_ * L_ * H_);  // barrier counter

  // Reset the barrier counter every call (graph replays don't re-poison ws).
  zero_counter<<<1, 1, 0, stream>>>(cnt);

  // Phase 1: x_proj = inputs @ Wi^T + bi   (M=65536, N=512, K=512)
  {
    const int nblocks = H_ / 64;                  // 8
    const int mblocks = (B_ * L_) / 32;           // 2048
    const int waves = nblocks * mblocks;          // 16384
    gemm_nt_bias<false><<<waves / 8, 256, 0, stream>>>(
        inputs, IN_, Wi, bi, xp, H_, B_ * L_, H_, IN_, nblocks);
  }

  // Phase 2: persistent recurrence, single kernel, Wh resident in LDS.
  {
    const size_t lds_bytes = (size_t)COLS_PER_WG_ * LDS_PITCH_ * sizeof(float);
    rnn_persistent<<<NWG_, WAVES_PER_WG_ * 32, lds_bytes, stream>>>(
        hidden, xp, Wh, bh, hn, cnt);
  }

  // Phase 3: output = h_n @ Wo^T + bo   (M=65536, N=1000, K=512)
  {
    const int nblocks = (C_ + 63) / 64;           // 16 (covers 1024, guarded)
    const int mblocks = (B_ * L_) / 32;           // 2048
    const int waves = nblocks * mblocks;          // 32768
    gemm_nt_bias<true><<<waves / 8, 256, 0, stream>>>(
        hn, H_, Wo, bo, out, C_, B_ * L_, C_, H_, nblocks);
  }
}